// ConsolidateBits_35888746725585
// MI455X (gfx1250) — compile-verified
//
#include <hip/hip_runtime.h>
#include <cstdint>

// ConsolidateBits: pack (x > 0.5) bits, 32 consecutive f32 -> one u32 word.
// Pure stream: 512MB in + 16MB out @ 23.3 TB/s -> ~22.7us floor.
//
// CDNA5 structure:
//  * GLOBAL_LOAD_ASYNC_TO_LDS_B128 stages each wave's 512B tile into LDS
//    (ASYNCcnt-tracked, zero VGPRs held for in-flight data, double-buffered
//    with s_wait_asynccnt — per-wave pipeline, no barriers needed).
//  * wave32 ballot: one v_cmp produces a full packed word (lane L -> bit L,
//    exactly the reference bit order).
//  * lane 0 stores 4 words per tile as one global_store_b128.

#define AS1 __attribute__((address_space(1)))
#define AS3 __attribute__((address_space(3)))

typedef int v4i __attribute__((__vector_size__(16)));  // b128 payload type

#if defined(__has_builtin)
#if __has_builtin(__builtin_amdgcn_global_load_async_to_lds_b128)
#define USE_ASYNC_BUILTIN 1
#endif
#if __has_builtin(__builtin_amdgcn_s_wait_asynccnt)
#define USE_WAIT_ASYNC_BUILTIN 1
#endif
#if __has_builtin(__builtin_amdgcn_ballot_w32)
#define USE_BALLOT_W32 1
#endif
#endif

__device__ __forceinline__ uint32_t ballot32(bool p) {
#if defined(USE_BALLOT_W32)
  return (uint32_t)__builtin_amdgcn_ballot_w32(p);
#else
  return (uint32_t)__ballot((int)p);
#endif
}

// Async-copy 16 bytes (per lane) from global to LDS. Wave-wide this copies a
// contiguous 512B tile. Tracked by ASYNCcnt, not LOADcnt.
__device__ __forceinline__ void async_copy16(const float* gsrc, uint8_t* ldst) {
#if defined(USE_ASYNC_BUILTIN)
  __builtin_amdgcn_global_load_async_to_lds_b128(
      (AS1 v4i*)gsrc, (AS3 v4i*)ldst, 0, 0);
#else
  uint32_t laddr = (uint32_t)(uintptr_t)(AS3 void*)ldst;  // raw LDS byte addr
  asm volatile("global_load_async_to_lds_b128 %0, %1, off"
               :: "v"(laddr), "v"(gsrc)
               : "memory");
#endif
}

__device__ __forceinline__ void wait_async_le(int n) {  // n is a literal 0/1
#if defined(USE_WAIT_ASYNC_BUILTIN)
  if (n == 0) __builtin_amdgcn_s_wait_asynccnt(0);
  else        __builtin_amdgcn_s_wait_asynccnt(1);
#else
  if (n == 0) asm volatile("s_wait_asynccnt 0x0" ::: "memory");
  else        asm volatile("s_wait_asynccnt 0x1" ::: "memory");
#endif
  asm volatile("" ::: "memory");  // compiler barrier: LDS was written by DMA
}

__global__ __launch_bounds__(256) void ConsolidateBits_pack_async(
    const float* __restrict__ x, uint32_t* __restrict__ out, int nWords) {
  // 8 waves/block * 2 buffers * 512B = 8KB LDS (out of 320KB/WGP).
  __shared__ uint8_t smem[8 * 2 * 512];

  const int lane   = (int)(threadIdx.x & 31u);
  const int wv     = (int)(threadIdx.x >> 5);
  const int wpb    = (int)(blockDim.x >> 5);
  const int stride = (int)gridDim.x * wpb;                 // waves in grid
  const int me     = (int)blockIdx.x * wpb + wv;
  const int nGroups = nWords >> 2;                          // 4 words per tile

  uint8_t* buf0 = smem + (size_t)wv * 1024u;
  uint8_t* buf1 = buf0 + 512u;

  // Prologue: stage up to two tiles (depth-2 pipeline).
  if (me < nGroups)
    async_copy16(x + (size_t)me * 128u + (size_t)lane * 4u,
                 buf0 + (size_t)lane * 16u);
  if (me + stride < nGroups)
    async_copy16(x + ((size_t)me + (size_t)stride) * 128u + (size_t)lane * 4u,
                 buf1 + (size_t)lane * 16u);

  int parity = 0;
  for (int g = me; g < nGroups; g += stride, parity ^= 1) {
    uint8_t* cur = parity ? buf1 : buf0;
    const bool more = (g + stride) < nGroups;   // is a newer tile in flight?

    // Oldest outstanding async load is this tile; wait for it to land.
    if (more) wait_async_le(1);
    else      wait_async_le(0);

    // Bank-conflict-free LDS reads: lane L hits bank (L + 32k) % 64.
    const float* f = (const float*)cur;
    float a = f[lane];
    float b = f[lane + 32];
    float c = f[lane + 64];
    float d = f[lane + 96];

    uint32_t m0 = ballot32(a > 0.5f);
    uint32_t m1 = ballot32(b > 0.5f);
    uint32_t m2 = ballot32(c > 0.5f);
    uint32_t m3 = ballot32(d > 0.5f);

    if (lane == 0) {
      uint4 w;
      w.x = m0; w.y = m1; w.z = m2; w.w = m3;
      *reinterpret_cast<uint4*>(out + (size_t)g * 4u) = w;  // store_b128
    }

    // Refill this buffer with the tile two steps ahead. ds reads above have
    // completed (their data fed the ballots) before this issues.
    int g2 = g + 2 * stride;
    if (g2 < nGroups)
      async_copy16(x + (size_t)g2 * 128u + (size_t)lane * 4u,
                   cur + (size_t)lane * 16u);
  }

  // Tail (nWords % 4 != 0): direct path, one word per wave-iteration.
  for (int wIdx = (nGroups << 2) + me; wIdx < nWords; wIdx += stride) {
    float v = x[(size_t)wIdx * 32u + (size_t)lane];
    uint32_t m = ballot32(v > 0.5f);
    if (lane == 0) out[wIdx] = m;
  }
}

extern "C" void kernel_launch(void* const* d_in, const int* in_sizes, int n_in,
                              void* d_out, int out_size, void* d_ws, size_t ws_size,
                              hipStream_t stream) {
  (void)in_sizes; (void)n_in; (void)d_ws; (void)ws_size;

  const float* x   = (const float*)d_in[0];
  uint32_t* out    = (uint32_t*)d_out;    // reference output dtype is uint32
  const int nWords = out_size;            // 4096 * (32768/32) = 4,194,304

  const int threads       = 256;          // 8 wave32s per block
  const int wavesPerBlock = threads / 32;
  const int nGroups       = nWords / 4;

  int blocks = (nGroups + wavesPerBlock - 1) / wavesPerBlock;
  if (blocks > 4096) blocks = 4096;       // grid-stride: ~32 tiles per wave
  if (blocks < 1)    blocks = 1;

  ConsolidateBits_pack_async<<<blocks, threads, 0, stream>>>(x, out, nWords);
}